// LIFNeuron_82901458748027
// MI455X (gfx1250) — compile-verified
//
#include <hip/hip_runtime.h>
#include <cstdint>
#include <cstddef>

// ---------------------------------------------------------------------------
// LIF spiking-neuron scan, shape [B=32, T=8, C=128, H=32, W=32] f32.
//   mem = 0.5*mem + x_t;  s = (mem > 1);  mem *= (1-s);  out = s
// Memory-bound (256 MiB @ 23.3 TB/s ~= 11 us). Strategy: one thread owns a
// float4 slot and runs the whole T=8 recurrence in registers; all 8 timestep
// tiles are prefetched with CDNA5 async global->LDS b128 loads (ASYNCcnt),
// consumed with progressive s_wait_asynccnt, stored non-temporally.
// ---------------------------------------------------------------------------

#define TPB      256
#define T_STEPS  8
#define SPV      32768u      // spatial float4-vectors per batch = 128*32*32/4
#define LOG2_SPV 15

typedef float v4f __attribute__((ext_vector_type(4)));

// The async-to-LDS builtin in this toolchain takes pointers to
// 'int __attribute__((vector_size(16)))' in AS1 (global) / AS3 (LDS).
typedef int v4i_vs __attribute__((vector_size(16)));
typedef __attribute__((address_space(1))) v4i_vs glb_v4i;
typedef __attribute__((address_space(3))) v4i_vs lds_v4i;

__device__ static __forceinline__ void async_load_b128(const float* g, const void* lds) {
#if __has_builtin(__builtin_amdgcn_global_load_async_to_lds_b128)
  // generic->AS1 keeps the 64-bit address; generic LDS pointer's low 32 bits
  // are the LDS byte offset (high bits are the shared aperture).
  __builtin_amdgcn_global_load_async_to_lds_b128(
      (glb_v4i*)(uintptr_t)g,
      (lds_v4i*)(uint32_t)(uintptr_t)lds,
      0, 0);
#else
  asm volatile("global_load_async_to_lds_b128 %0, %1, off"
               :
               : "v"((uint32_t)(uintptr_t)lds), "v"((uint64_t)(uintptr_t)g)
               : "memory");
#endif
}

template <int N>
__device__ static __forceinline__ void wait_asynccnt() {
#if __has_builtin(__builtin_amdgcn_s_wait_asynccnt)
  __builtin_amdgcn_s_wait_asynccnt(N);
#else
  asm volatile("s_wait_asynccnt %0" ::"i"(N) : "memory");
#endif
}

template <int TI>
__device__ static __forceinline__ void lif_step(v4f& mem, const v4f* lds_buf, uint32_t tid,
                                                float* gout, size_t elems_per_t) {
  // async loads complete in order: waiting ASYNCcnt <= (T-1-TI) guarantees
  // buffers 0..TI are resident in LDS.
  wait_asynccnt<T_STEPS - 1 - TI>();
  v4f xv = lds_buf[TI * TPB + tid];   // ds_load_b128, conflict-free
  v4f m, s;
#pragma unroll
  for (int i = 0; i < 4; ++i) m[i] = fmaf(0.5f, mem[i], xv[i]);
#pragma unroll
  for (int i = 0; i < 4; ++i) {
    const bool fire = m[i] > 1.0f;    // heaviside(m - v_th), v_th = 1
    s[i]   = fire ? 1.0f : 0.0f;
    mem[i] = fire ? 0.0f : m[i];      // m * (1 - s), branchless
  }
  __builtin_nontemporal_store(s, (v4f*)(gout + (size_t)TI * elems_per_t));
}

__global__ __launch_bounds__(TPB) void lif_spikes_kernel(const float* __restrict__ x,
                                                         float* __restrict__ out,
                                                         uint32_t n_vec) {
  __shared__ v4f lds_buf[T_STEPS * TPB];   // 32 KiB: 8 timestep tiles per block

  const uint32_t tid = threadIdx.x;
  const uint32_t v   = blockIdx.x * TPB + tid;
  if (v >= n_vec) return;

  const uint32_t b  = v >> LOG2_SPV;        // batch index
  const uint32_t sp = v & (SPV - 1u);       // float4 index within C*H*W
  const size_t elems_per_t = (size_t)SPV * 4;                      // 131072 floats
  const size_t base = (size_t)b * T_STEPS * elems_per_t + (size_t)sp * 4;

  const float* gin  = x + base;
  float*       gout = out + base;

  // Kick off the entire time series: 8 in-flight async b128 loads per thread.
#pragma unroll
  for (int t = 0; t < T_STEPS; ++t)
    async_load_b128(gin + (size_t)t * elems_per_t, &lds_buf[t * TPB + tid]);

  v4f mem = {0.0f, 0.0f, 0.0f, 0.0f};
  lif_step<0>(mem, lds_buf, tid, gout, elems_per_t);
  lif_step<1>(mem, lds_buf, tid, gout, elems_per_t);
  lif_step<2>(mem, lds_buf, tid, gout, elems_per_t);
  lif_step<3>(mem, lds_buf, tid, gout, elems_per_t);
  lif_step<4>(mem, lds_buf, tid, gout, elems_per_t);
  lif_step<5>(mem, lds_buf, tid, gout, elems_per_t);
  lif_step<6>(mem, lds_buf, tid, gout, elems_per_t);
  lif_step<7>(mem, lds_buf, tid, gout, elems_per_t);
}

extern "C" void kernel_launch(void* const* d_in, const int* in_sizes, int n_in,
                              void* d_out, int out_size, void* d_ws, size_t ws_size,
                              hipStream_t stream) {
  (void)n_in; (void)out_size; (void)d_ws; (void)ws_size;
  const float* x   = (const float*)d_in[0];
  float*       out = (float*)d_out;

  // Reference shape [32, 8, 128, 32, 32]: n = 33,554,432 floats.
  // Each thread handles T=8 steps of one float4 slot -> n / 32 threads.
  const uint32_t n     = (uint32_t)in_sizes[0];
  const uint32_t n_vec = n / (T_STEPS * 4);
  const uint32_t grid  = (n_vec + TPB - 1) / TPB;

  lif_spikes_kernel<<<grid, TPB, 0, stream>>>(x, out, n_vec);
}